// SRLGCNModel_84842783965237
// MI455X (gfx1250) — compile-verified
//
#include <hip/hip_runtime.h>
#include <hip/hip_bf16.h>

typedef __attribute__((ext_vector_type(2))) float v2f;
typedef __attribute__((ext_vector_type(8))) float v8f;

#define HID 64
#define NCLS 16
#define NGRAPH 64

// ---------------------------------------------------------------------------
// degree / norm prep
// ---------------------------------------------------------------------------
__global__ void k_deg_init(float* __restrict__ deg, int N) {
    int i = blockIdx.x * blockDim.x + threadIdx.x;
    if (i < N) deg[i] = 1.0f;                       // self-loop weight
}

__global__ void k_deg_scatter(const int* __restrict__ dst, const float* __restrict__ ew,
                              float* __restrict__ deg, int E) {
    int e = blockIdx.x * blockDim.x + threadIdx.x;
    if (e < E) unsafeAtomicAdd(&deg[dst[e]], ew[e]);
}

__global__ void k_dis(const float* __restrict__ deg, float* __restrict__ dis, int N) {
    int i = blockIdx.x * blockDim.x + threadIdx.x;
    if (i < N) {
        float d = deg[i];
        dis[i] = (d > 0.0f) ? rsqrtf(fmaxf(d, 1e-30f)) : 0.0f;
    }
}

__global__ void k_norm(const int* __restrict__ src, const int* __restrict__ dst,
                       const float* __restrict__ ew, const float* __restrict__ dis,
                       float* __restrict__ nrm, int E) {
    int e = blockIdx.x * blockDim.x + threadIdx.x;
    if (e < E) nrm[e] = dis[src[e]] * ew[e] * dis[dst[e]];
}

// ---------------------------------------------------------------------------
// fp32 WMMA GEMM:  H[M,64] = X[M,K] @ W[K,64],  K compile-time (64 or 128).
// Block = 128 threads = 4 waves; one shared A-tile (16 rows of X) in LDS,
// wave w computes the 16x16 tile at columns [16w, 16w+16).
// B is read per-lane from global (W is <=32KB, L2-resident); loop fully
// unrolled so loads are hoisted ahead of the v_wmma chain.
// ---------------------------------------------------------------------------
template<int K>
__global__ __launch_bounds__(128) void k_gemm_wmma(const float* __restrict__ X,
                                                   const float* __restrict__ W,
                                                   float* __restrict__ H, int M) {
    constexpr int AS = K + 4;                 // padded stride -> conflict-free ds_load_b64
    __shared__ float As[16 * AS];

    const int tid  = threadIdx.x;             // 0..127
    const int wave = tid >> 5;                // 0..3 -> N tile
    const int lane = tid & 31;
    const int row0 = blockIdx.x * 16;
    const int n0   = wave * 16;

    // stage shared A tile, coalesced along K
    for (int idx = tid; idx < 16 * K; idx += 128) {
        int r = idx / K, c = idx % K;         // K is constexpr pow2 -> shifts
        int row = row0 + r; if (row >= M) row = M - 1;
        As[r * AS + c] = X[(size_t)row * K + c];
    }
    __syncthreads();

    const int half = lane >> 4;               // 0: K-pair {0,1}; 1: K-pair {2,3}
    const int l    = lane & 15;
    const float* __restrict__ Wp = W + (size_t)(2 * half) * HID + n0 + l;

    v8f acc = {};
#pragma unroll
    for (int q = 0; q < K / 4; ++q) {
        v2f a, b;
        a.x = As[l * AS + 4 * q + 2 * half];  // merges into ds_load_b64
        a.y = As[l * AS + 4 * q + 2 * half + 1];
        b.x = Wp[(size_t)(4 * q) * HID];      // row 4q + 2*half
        b.y = Wp[(size_t)(4 * q + 1) * HID];  // row 4q + 2*half + 1
        acc = __builtin_amdgcn_wmma_f32_16x16x4_f32(false, a, false, b,
                                                    (short)0, acc, false, false);
    }

    // D layout: VGPR r -> row (r + 8*half), lane -> column (n0 + l)
    if (row0 + 16 <= M) {                     // fast path: no per-row guards
        float* __restrict__ Hp = H + (size_t)(row0 + 8 * half) * HID + n0 + l;
#pragma unroll
        for (int r = 0; r < 8; ++r) Hp[(size_t)r * HID] = acc[r];
    } else {
#pragma unroll
        for (int r = 0; r < 8; ++r) {
            int row = row0 + r + 8 * half;
            if (row < M) H[(size_t)row * HID + n0 + l] = acc[r];
        }
    }
}

// ---------------------------------------------------------------------------
// aggregation: self-loop init, edge scatter, bias+relu
// ---------------------------------------------------------------------------
__global__ void k_agg_init(const float* __restrict__ Hm, const float* __restrict__ dis,
                           float* __restrict__ A, int N) {
    int t = blockIdx.x * blockDim.x + threadIdx.x;
    if (t < N * HID) {
        int n = t >> 6;
        float s = dis[n];
        A[t] = Hm[t] * s * s;
    }
}

__global__ void k_scatter(const int* __restrict__ src, const int* __restrict__ dst,
                          const float* __restrict__ nrm, const float* __restrict__ Hm,
                          float* __restrict__ A, int E) {
    int t = blockIdx.x * blockDim.x + threadIdx.x;
    if (t < E * HID) {
        int e = t >> 6, c = t & 63;
        int s = src[e], d = dst[e];
        unsafeAtomicAdd(&A[(size_t)d * HID + c], Hm[(size_t)s * HID + c] * nrm[e]);
    }
}

__global__ void k_bias_relu(float* __restrict__ A, const float* __restrict__ b, int N) {
    int t = blockIdx.x * blockDim.x + threadIdx.x;
    if (t < N * HID) A[t] = fmaxf(A[t] + b[t & 63], 0.0f);
}

// ---------------------------------------------------------------------------
// pooling
// ---------------------------------------------------------------------------
__global__ void k_pool_zero(float* __restrict__ pooled, float* __restrict__ cnt) {
    int t = blockIdx.x * blockDim.x + threadIdx.x;
    if (t < NGRAPH * HID) pooled[t] = 0.0f;
    if (t < NGRAPH) cnt[t] = 0.0f;
}

__global__ __launch_bounds__(256) void k_pool(const float* __restrict__ A,
                                              const int* __restrict__ batch,
                                              float* __restrict__ pooled,
                                              float* __restrict__ cnt, int N) {
    __shared__ float s_sum[NGRAPH * HID];   // 16 KB
    __shared__ float s_cnt[NGRAPH];
    for (int i = threadIdx.x; i < NGRAPH * HID; i += 256) s_sum[i] = 0.0f;
    if (threadIdx.x < NGRAPH) s_cnt[threadIdx.x] = 0.0f;
    __syncthreads();

    const long long total  = (long long)N * HID;
    const long long stride = (long long)gridDim.x * blockDim.x;
    for (long long t = (long long)blockIdx.x * blockDim.x + threadIdx.x; t < total; t += stride) {
        int n = (int)(t >> 6), c = (int)(t & 63);
        int g = batch[n];
        unsafeAtomicAdd(&s_sum[g * HID + c], A[t]);
        if (c == 0) unsafeAtomicAdd(&s_cnt[g], 1.0f);
    }
    __syncthreads();
    for (int i = threadIdx.x; i < NGRAPH * HID; i += 256) {
        float v = s_sum[i];
        if (v != 0.0f) unsafeAtomicAdd(&pooled[i], v);
    }
    if (threadIdx.x < NGRAPH) {
        float v = s_cnt[threadIdx.x];
        if (v != 0.0f) unsafeAtomicAdd(&cnt[threadIdx.x], v);
    }
}

__global__ void k_head(const float* __restrict__ pooled, const float* __restrict__ cnt,
                       const float* __restrict__ Wl, const float* __restrict__ bl,
                       float* __restrict__ out) {
    int t = blockIdx.x * blockDim.x + threadIdx.x;
    if (t >= NGRAPH * NCLS) return;
    int g = t >> 4, k = t & 15;
    float inv = 1.0f / fmaxf(cnt[g], 1.0f);
    float s = 0.0f;
    for (int c = 0; c < HID; ++c) s += pooled[g * HID + c] * inv * Wl[c * NCLS + k];
    out[t] = s + bl[k];
}

// ---------------------------------------------------------------------------
extern "C" void kernel_launch(void* const* d_in, const int* in_sizes, int n_in,
                              void* d_out, int out_size, void* d_ws, size_t ws_size,
                              hipStream_t stream) {
    const float* x   = (const float*)d_in[0];
    const int*   ei  = (const int*)  d_in[1];
    const float* ew  = (const float*)d_in[2];
    const int*   bat = (const int*)  d_in[3];
    const float* W1  = (const float*)d_in[4];
    const float* b1  = (const float*)d_in[5];
    const float* W2  = (const float*)d_in[6];
    const float* b2  = (const float*)d_in[7];
    const float* Wl  = (const float*)d_in[8];
    const float* bl  = (const float*)d_in[9];
    float* out = (float*)d_out;

    const int IN_CH = 128;
    const int N = in_sizes[0] / IN_CH;
    const int E = in_sizes[1] / 2;
    const int* srcI = ei;
    const int* dstI = ei + E;

    // workspace layout (floats)
    float* ws     = (float*)d_ws;
    float* deg    = ws;                 ws += N;
    float* dis    = ws;                 ws += N;
    float* nrm    = ws;                 ws += E;
    float* h      = ws;                 ws += (size_t)N * HID;
    float* a      = ws;                 ws += (size_t)N * HID;
    float* pooled = ws;                 ws += NGRAPH * HID;
    float* cnt    = ws;                 ws += NGRAPH;

    const int B = 256;
    const int gN  = (N + B - 1) / B;
    const int gE  = (E + B - 1) / B;
    const int gNC = (N * HID + B - 1) / B;
    const int gEC = (E * HID + B - 1) / B;

    // normalization coefficients
    k_deg_init   <<<gN, B, 0, stream>>>(deg, N);
    k_deg_scatter<<<gE, B, 0, stream>>>(dstI, ew, deg, E);
    k_dis        <<<gN, B, 0, stream>>>(deg, dis, N);
    k_norm       <<<gE, B, 0, stream>>>(srcI, dstI, ew, dis, nrm, E);

    dim3 gemmGrid((N + 15) / 16);   // 4 waves/block cover the 4 N-tiles

    // ---- layer 1: h = x @ W1 ; a = scatter(norm * h) ; a = relu(a + b1)
    k_gemm_wmma<128><<<gemmGrid, 128, 0, stream>>>(x, W1, h, N);
    k_agg_init <<<gNC, B, 0, stream>>>(h, dis, a, N);
    k_scatter  <<<gEC, B, 0, stream>>>(srcI, dstI, nrm, h, a, E);
    k_bias_relu<<<gNC, B, 0, stream>>>(a, b1, N);

    // ---- layer 2: h = a @ W2 ; a = scatter(norm * h) ; a = relu(a + b2)
    k_gemm_wmma<64><<<gemmGrid, 128, 0, stream>>>(a, W2, h, N);
    k_agg_init <<<gNC, B, 0, stream>>>(h, dis, a, N);           // a overwritten after gemm read
    k_scatter  <<<gEC, B, 0, stream>>>(srcI, dstI, nrm, h, a, E);
    k_bias_relu<<<gNC, B, 0, stream>>>(a, b2, N);

    // ---- mean pool + head
    k_pool_zero<<<(NGRAPH * HID + B - 1) / B, B, 0, stream>>>(pooled, cnt);
    int poolBlocks = gNC < 2048 ? gNC : 2048;
    k_pool<<<poolBlocks, B, 0, stream>>>(a, bat, pooled, cnt, N);
    k_head<<<(NGRAPH * NCLS + B - 1) / B, B, 0, stream>>>(pooled, cnt, Wl, bl, out);
}